// GCN_5660766896358
// MI455X (gfx1250) — compile-verified
//
#include <hip/hip_runtime.h>
#include <hip/hip_bf16.h>

// ---------------------------------------------------------------------------
// GCNConv forward for MI455X (gfx1250, wave32, WMMA).
//   h   = x @ W                      (fp32 WMMA 16x16x4)
//   deg = in-degree(+self), dinv = deg^{-1/2}
//   out = sum_{e:(s->d)} dinv[s]*dinv[d]*h[s]  + dinv[d]^2*h[d]   (L2 atomics)
//   out = log_softmax(out + b)       (one wave per row)
// ---------------------------------------------------------------------------

typedef float v2f __attribute__((ext_vector_type(2)));
typedef float v8f __attribute__((ext_vector_type(8)));

#define F_IN 512
#define C_OUT 64

// ---------------- degree / normalization ----------------------------------

__global__ void gcn_deg_init(float* deg, int N) {
  int i = blockIdx.x * blockDim.x + threadIdx.x;
  if (i < N) deg[i] = 1.0f;  // self loop
}

__global__ void gcn_deg_count(const int* __restrict__ dst, float* deg, int E) {
  int stride = gridDim.x * blockDim.x;
  for (int e = blockIdx.x * blockDim.x + threadIdx.x; e < E; e += stride)
    atomicAdd(&deg[dst[e]], 1.0f);
}

__global__ void gcn_deg_to_dinv(float* deg, int N) {
  int i = blockIdx.x * blockDim.x + threadIdx.x;
  if (i < N) deg[i] = rsqrtf(deg[i]);  // deg >= 1 always (self loop)
}

// ---------------- projection: h = x @ W via V_WMMA_F32_16X16X4_F32 ---------
// One wave per 16-row tile; 4 accumulators cover all 64 output channels.
// A frag: lane (16*half + m) holds x[row0+m, k+2*half .. k+2*half+1].
// B frag: lane (16*half + n) holds W[k+2*half, n0+n], W[k+2*half+1, n0+n].
// D:      VGPR v of lane (16*half + n) is element (M = v + 8*half, N = n).

__global__ __launch_bounds__(128) void gcn_gemm_wmma(
    const float* __restrict__ x, const float* __restrict__ W,
    float* __restrict__ h, int N) {
  const int lane = threadIdx.x & 31;
  const int wid  = threadIdx.x >> 5;
  const int tile = blockIdx.x * 4 + wid;
  if (tile * 16 >= N) return;  // wave-uniform: EXEC stays all-ones for WMMA

  const int half = lane >> 4;  // 0 or 1
  const int l16  = lane & 15;
  const int row0 = tile * 16;

  const float* arow = x + (size_t)(row0 + l16) * F_IN + 2 * half;
  const float* bbas = W + (size_t)(2 * half) * C_OUT + l16;

  v8f acc0 = {}, acc1 = {}, acc2 = {}, acc3 = {};

  #pragma unroll 4
  for (int k = 0; k < F_IN; k += 4) {
    v2f a = *(const v2f*)(arow + k);
    const float* bk = bbas + (size_t)k * C_OUT;
    v2f b0 = { bk[0],  bk[0  + C_OUT] };
    v2f b1 = { bk[16], bk[16 + C_OUT] };
    v2f b2 = { bk[32], bk[32 + C_OUT] };
    v2f b3 = { bk[48], bk[48 + C_OUT] };
    acc0 = __builtin_amdgcn_wmma_f32_16x16x4_f32(false, a, false, b0, (short)0, acc0, false, false);
    acc1 = __builtin_amdgcn_wmma_f32_16x16x4_f32(false, a, false, b1, (short)0, acc1, false, false);
    acc2 = __builtin_amdgcn_wmma_f32_16x16x4_f32(false, a, false, b2, (short)0, acc2, false, false);
    acc3 = __builtin_amdgcn_wmma_f32_16x16x4_f32(false, a, false, b3, (short)0, acc3, false, false);
  }

  float* hp = h + (size_t)row0 * C_OUT;
  #pragma unroll
  for (int v = 0; v < 8; ++v) {
    const size_t r = (size_t)(v + 8 * half) * C_OUT;
    hp[r + 0  + l16] = acc0[v];
    hp[r + 16 + l16] = acc1[v];
    hp[r + 32 + l16] = acc2[v];
    hp[r + 48 + l16] = acc3[v];
  }
}

// ---------------- self-loop term (also deterministically inits d_out) ------

__global__ void gcn_self_init(const float* __restrict__ h,
                              const float* __restrict__ dinv,
                              float* __restrict__ out, int N) {
  int stride = gridDim.x * blockDim.x;
  int total = N * C_OUT;
  for (int i = blockIdx.x * blockDim.x + threadIdx.x; i < total; i += stride) {
    float di = dinv[i >> 6];  // i / 64
    out[i] = h[i] * di * di;
  }
}

// ---------------- edge aggregation: out[dst] += norm * h[src] --------------
// One (edge, float4-group) pair per logical work item; a wave covers 2 edges
// with fully coalesced 256B reads of h[src]. Atomics land in L2 (out = 25.6MB
// << 192MB L2).

__global__ void gcn_aggregate(const int* __restrict__ src,
                              const int* __restrict__ dst,
                              const float* __restrict__ h,
                              const float* __restrict__ dinv,
                              float* __restrict__ out, int E) {
  long long stride = (long long)gridDim.x * blockDim.x;
  long long total  = (long long)E * 16;  // 16 float4 groups per edge
  for (long long i = (long long)blockIdx.x * blockDim.x + threadIdx.x;
       i < total; i += stride) {
    int e = (int)(i >> 4);
    int g = (int)(i & 15);
    int s = src[e];
    int d = dst[e];
    float norm = dinv[s] * dinv[d];
    const float4 hv = *(const float4*)(h + (size_t)s * C_OUT + g * 4);
    float* o = out + (size_t)d * C_OUT + g * 4;
    atomicAdd(o + 0, hv.x * norm);
    atomicAdd(o + 1, hv.y * norm);
    atomicAdd(o + 2, hv.z * norm);
    atomicAdd(o + 3, hv.w * norm);
  }
}

// ---------------- bias + log_softmax (one wave32 per row, in place) --------

__global__ void gcn_bias_logsoftmax(float* __restrict__ out,
                                    const float* __restrict__ bias, int N) {
  int lane = threadIdx.x & 31;
  int row  = blockIdx.x * (blockDim.x >> 5) + (threadIdx.x >> 5);
  if (row >= N) return;
  float* o = out + (size_t)row * C_OUT;
  float v0 = o[lane]      + bias[lane];
  float v1 = o[lane + 32] + bias[lane + 32];
  float m = fmaxf(v0, v1);
  #pragma unroll
  for (int off = 16; off > 0; off >>= 1) m = fmaxf(m, __shfl_xor(m, off, 32));
  float s = __expf(v0 - m) + __expf(v1 - m);
  #pragma unroll
  for (int off = 16; off > 0; off >>= 1) s += __shfl_xor(s, off, 32);
  float lse = __logf(s);
  o[lane]      = v0 - m - lse;
  o[lane + 32] = v1 - m - lse;
}

// ---------------------------------------------------------------------------

extern "C" void kernel_launch(void* const* d_in, const int* in_sizes, int n_in,
                              void* d_out, int out_size, void* d_ws, size_t ws_size,
                              hipStream_t stream) {
  const float* x  = (const float*)d_in[0];
  const float* W  = (const float*)d_in[1];
  const float* b  = (const float*)d_in[2];
  const int*   ei = (const int*)d_in[3];

  const int N = in_sizes[0] / F_IN;
  const int E = in_sizes[3] / 2;
  const int* src = ei;       // edge_index[0]
  const int* dst = ei + E;   // edge_index[1]
  float* out = (float*)d_out;

  // workspace: dinv [N] f32, then h [N*64] f32
  char*  ws   = (char*)d_ws;
  float* dinv = (float*)ws;
  size_t off  = ((size_t)N * sizeof(float) + 255) & ~(size_t)255;
  float* h    = (float*)(ws + off);

  const int T = 256;

  // 1. degree (with self loop) -> dinv = deg^{-1/2}
  gcn_deg_init<<<(N + T - 1) / T, T, 0, stream>>>(dinv, N);
  {
    int blocks = (E + T - 1) / T;
    if (blocks > 8192) blocks = 8192;
    gcn_deg_count<<<blocks, T, 0, stream>>>(dst, dinv, E);
  }
  gcn_deg_to_dinv<<<(N + T - 1) / T, T, 0, stream>>>(dinv, N);

  // 2. h = x @ W  (fp32 WMMA; 4 waves = 4 row-tiles per block)
  {
    int tiles  = (N + 15) / 16;
    int blocks = (tiles + 3) / 4;
    gcn_gemm_wmma<<<blocks, 128, 0, stream>>>(x, W, h, N);
  }

  // 3. out = dinv^2 * h (self loops; also re-initializes d_out every call)
  {
    int total  = N * C_OUT;
    int blocks = (total + T - 1) / T;
    if (blocks > 16384) blocks = 16384;
    gcn_self_init<<<blocks, T, 0, stream>>>(h, dinv, out, N);
  }

  // 4. scatter-add edge messages
  {
    long long total = (long long)E * 16;
    long long blk   = (total + T - 1) / T;
    int blocks = blk > 16384 ? 16384 : (int)blk;
    gcn_aggregate<<<blocks, T, 0, stream>>>(src, dst, h, dinv, out, E);
  }

  // 5. bias + log_softmax, one wave per row (8 rows per 256-thread block)
  {
    int rows_per_block = T / 32;
    int blocks = (N + rows_per_block - 1) / rows_per_block;
    gcn_bias_logsoftmax<<<blocks, T, 0, stream>>>(out, b, N);
  }
}